// LSHSelfAttention_16501264351598
// MI455X (gfx1250) — compile-verified
//
#include <hip/hip_runtime.h>
#include <hip/hip_bf16.h>
#include <math.h>

// ---------------------------------------------------------------------------
// Problem constants (Reformer LSH attention)
// ---------------------------------------------------------------------------
#define BATCH      2
#define SEQ        4096
#define EMB        768
#define HEADS      8
#define DH         64
#define QKV        (HEADS * DH)     // 512
#define BT         (BATCH * SEQ)    // 8192
#define NHASH      4
#define NBUCK      32               // SEQ / BUCKET
#define BUCKET     128
#define NKEY       (NHASH * SEQ)    // 16384 keys per bh
#define BHN        (BATCH * HEADS)  // 16
#define NCHUNK     (NHASH * NBUCK)  // 128 chunks per bh
#define SELF_VAL   (-5e4f)

typedef float v2f __attribute__((ext_vector_type(2)));
typedef float v8f __attribute__((ext_vector_type(8)));

// f32 WMMA fragment layout (CDNA5 ISA 7.12.2, wave32):
//   A (16x4):  lane L<16 -> M=L, K={0,1}; lane L>=16 -> M=L-16, K={2,3}
//   B (4x16):  lane L<16 -> N=L, K={0,1}; lane L>=16 -> N=L-16, K={2,3}
//   C/D (16x16): vgpr i -> M = i + 8*(lane>=16), N = lane%16
//
// All LDS tiles are laid out so a fragment is ONE aligned ds_load_b64:
// the two K values a lane needs are contiguous, and the per-lane row
// stride is ≡ 4 (mod 32) so the 32 lanes' 64 words hit all 64 banks once.

// ---------------------------------------------------------------------------
// Generic f32 WMMA GEMM:  C(MxN) = A(MxK) @ B(KxN) (+ bias)
// Block tile 128x64, 8 waves; wave w owns rows [16w,16w+16) x 64 cols.
//   As2[m][36] : A row-major, K pair contiguous, stride 36 ≡ 4 (mod 32)
//   BsT[n][36] : B column-major (N-major), K pair contiguous
// ---------------------------------------------------------------------------
#define GEMM_LDK 36

__global__ __launch_bounds__(256) void wmma_gemm_f32_kernel(
    const float* __restrict__ A, const float* __restrict__ B,
    const float* __restrict__ bias, float* __restrict__ C,
    int M, int N, int K) {
  __shared__ float As2[128 * GEMM_LDK];   // [m][k]
  __shared__ float BsT[64 * GEMM_LDK];    // [n][k]

  const int tid  = threadIdx.x;
  const int bm   = blockIdx.y * 128;
  const int bn   = blockIdx.x * 64;
  const int wave = tid >> 5;
  const int lane = tid & 31;          // wave32
  const int half = lane >> 4;
  const int r    = lane & 15;
  const int m0   = wave * 16;

  v8f acc[4] = {v8f{}, v8f{}, v8f{}, v8f{}};

  for (int k0 = 0; k0 < K; k0 += 32) {
    // --- stage A tile (128x32): row-major, 16B-aligned float4 stores ---
    {
      const int row = tid >> 1;
      const int ks  = (tid & 1) * 16;
      const float4* src = (const float4*)(A + (size_t)(bm + row) * K + k0 + ks);
      if (k0 + 32 < K)  // CDNA5 global_prefetch_b8 for next tile
        __builtin_prefetch(A + (size_t)(bm + row) * K + k0 + 32 + ks, 0, 0);
#pragma unroll
      for (int q = 0; q < 4; ++q)
        *(float4*)&As2[row * GEMM_LDK + ks + q * 4] = src[q];
    }
    // --- stage B tile (32x64) transposed into BsT[n][k] ---
    {
      const int krow = tid >> 3;
      const int nseg = (tid & 7) * 8;
      const float4* src = (const float4*)(B + (size_t)(k0 + krow) * N + bn + nseg);
      float4 v0 = src[0], v1 = src[1];
      BsT[(nseg + 0) * GEMM_LDK + krow] = v0.x;
      BsT[(nseg + 1) * GEMM_LDK + krow] = v0.y;
      BsT[(nseg + 2) * GEMM_LDK + krow] = v0.z;
      BsT[(nseg + 3) * GEMM_LDK + krow] = v0.w;
      BsT[(nseg + 4) * GEMM_LDK + krow] = v1.x;
      BsT[(nseg + 5) * GEMM_LDK + krow] = v1.y;
      BsT[(nseg + 6) * GEMM_LDK + krow] = v1.z;
      BsT[(nseg + 7) * GEMM_LDK + krow] = v1.w;
    }
    __syncthreads();

#pragma unroll
    for (int kt = 0; kt < 8; ++kt) {
      const int kk = kt * 4 + half * 2;
      const v2f a = *(const v2f*)&As2[(m0 + r) * GEMM_LDK + kk];
#pragma unroll
      for (int n = 0; n < 4; ++n) {
        const v2f bf = *(const v2f*)&BsT[(n * 16 + r) * GEMM_LDK + kk];
        acc[n] = __builtin_amdgcn_wmma_f32_16x16x4_f32(
            false, a, false, bf, (short)0, acc[n], false, false);
      }
    }
    __syncthreads();
  }

#pragma unroll
  for (int n = 0; n < 4; ++n) {
    const int col = bn + n * 16 + r;
    const float bb = bias ? bias[col] : 0.0f;
#pragma unroll
    for (int i = 0; i < 8; ++i) {
      const int row = bm + m0 + half * 8 + i;
      C[(size_t)row * N + col] = acc[n][i] + bb;
    }
  }
}

// ---------------------------------------------------------------------------
// LSH hashing: buckets = argmax(concat(R, -R)), key = bucket*SEQ + pos.
// key is injective: pos = key & 4095, bucket = key >> 12, hash = key >> 17.
// ---------------------------------------------------------------------------
__global__ __launch_bounds__(256) void lsh_hash_kernel(
    const float* __restrict__ qkb,        // (BT, QKV)
    const float* __restrict__ rot,        // (DH, NHASH, 16) flat
    unsigned* __restrict__ keys) {        // (BHN, NKEY)
  __shared__ float rs[DH * NHASH * 16];   // 16 KB
  const int tid = threadIdx.x;
  for (int i = tid; i < DH * NHASH * 16; i += 256) rs[i] = rot[i];
  __syncthreads();

  const int g  = blockIdx.x * 256 + tid;  // 0 .. BHN*SEQ-1
  const int bh = g >> 12;
  const int t  = g & (SEQ - 1);
  const int b_ = bh >> 3;
  const int h  = bh & 7;

  const float* qrow = qkb + ((size_t)b_ * SEQ + t) * QKV + h * DH;
  float q[DH];
#pragma unroll
  for (int f = 0; f < DH; ++f) q[f] = qrow[f];

  for (int hs = 0; hs < NHASH; ++hs) {
    float rv[16];
#pragma unroll
    for (int i = 0; i < 16; ++i) {
      float s = 0.0f;
#pragma unroll
      for (int f = 0; f < DH; ++f) s += q[f] * rs[f * (NHASH * 16) + hs * 16 + i];
      rv[i] = s;
    }
    float best = rv[0];
    int bidx = 0;
#pragma unroll
    for (int i = 1; i < 16; ++i) if (rv[i] > best) { best = rv[i]; bidx = i; }
#pragma unroll
    for (int i = 0; i < 16; ++i) if (-rv[i] > best) { best = -rv[i]; bidx = i + 16; }
    const unsigned bucket = (unsigned)(bidx + hs * NBUCK);
    keys[(size_t)bh * NKEY + hs * SEQ + t] = bucket * (unsigned)SEQ + (unsigned)t;
  }
}

// ---------------------------------------------------------------------------
// In-LDS bitonic sort of 16384 u32 keys, one workgroup per bh (64 KB LDS).
// ---------------------------------------------------------------------------
__global__ __launch_bounds__(1024) void bitonic_sort_kernel(unsigned* __restrict__ keys) {
  __shared__ unsigned sk[NKEY];
  const int tid = threadIdx.x;
  unsigned* gk = keys + (size_t)blockIdx.x * NKEY;
  for (int i = tid; i < NKEY; i += 1024) sk[i] = gk[i];
  __syncthreads();

  for (unsigned k = 2; k <= NKEY; k <<= 1) {
    for (unsigned j = k >> 1; j > 0; j >>= 1) {
      for (unsigned i = tid; i < NKEY; i += 1024) {
        const unsigned ixj = i ^ j;
        if (ixj > i) {
          const unsigned a = sk[i], b = sk[ixj];
          const bool up = ((i & k) == 0);
          if (up ? (a > b) : (a < b)) { sk[i] = b; sk[ixj] = a; }
        }
      }
      __syncthreads();
    }
  }
  for (int i = tid; i < NKEY; i += 1024) gk[i] = sk[i];
}

// ---------------------------------------------------------------------------
// Chunk attention. One workgroup per (bh, chunk). 8 waves.
// LDS layout (~306.7 KB < 320 KB/WG), fragment-facing strides ≡ 4 (mod 64):
//   bq  [128][68]  : q rows, K(=feature) pair contiguous (A of dots)
//   bk  [256][68]  : normalized k rows, N-major for B of dots
//   bvF [64][260]  : v feature-major, K(=kv row) pair contiguous (B of bo)
//   dots[128][260] : scores / probs (A of bo)
// ---------------------------------------------------------------------------
#define BQ_LD   68
#define BK_LD   68
#define BV_LD   260
#define DOTS_LD 260

__global__ __launch_bounds__(256) void lsh_attn_chunk_kernel(
    const float* __restrict__ qkb,        // (BT, QKV)
    const float* __restrict__ vb,         // (BT, QKV)
    const unsigned* __restrict__ keys,    // sorted, (BHN, NKEY)
    float* __restrict__ o_hash,           // (BHN, NKEY, DH) scattered by ticker
    float* __restrict__ lse_out) {        // (BHN, NKEY)
  extern __shared__ float smem[];
  float* bq   = smem;                           // 128*68  = 8704
  float* bk   = bq  + 128 * BQ_LD;              // 256*68  = 17408
  float* bvF  = bk  + 256 * BK_LD;              // 64*260  = 16640
  float* dots = bvF + 64 * BV_LD;               // 128*260 = 33280
  int*   pos_q  = (int*)(dots + 128 * DOTS_LD); // 128
  int*   pos_kv = pos_q + 128;                  // 256
  int*   tick_q = pos_kv + 256;                 // 128
  float* lse_s  = (float*)(tick_q + 128);       // 128

  const int tid = threadIdx.x;
  const int bh  = blockIdx.x >> 7;
  const int c   = blockIdx.x & (NCHUNK - 1);
  const int b_  = bh >> 3;
  const int h   = bh & 7;
  const unsigned* skeys = keys + (size_t)bh * NKEY;

  // ---- gather rows: j<128 current chunk (also q rows), j>=128 previous chunk
  {
    const int j  = tid;                       // blockDim == 256
    const int cc = (j < 128) ? c : ((c + NCHUNK - 1) & (NCHUNK - 1));
    const int rr = j & 127;
    const unsigned key = skeys[cc * BUCKET + rr];
    const int pos  = (int)(key & (SEQ - 1));
    pos_kv[j] = pos;
    const float* qrow = qkb + ((size_t)b_ * SEQ + pos) * QKV + h * DH;
    const float* vrow = vb  + ((size_t)b_ * SEQ + pos) * QKV + h * DH;
    float tmp[DH];
    float ss = 0.0f;
#pragma unroll
    for (int f = 0; f < DH; ++f) { float x = qrow[f]; tmp[f] = x; ss += x * x; }
    const float sc = 1.0f / fmaxf(sqrtf(ss), 1e-12f);
#pragma unroll
    for (int f = 0; f < DH; ++f) bk[j * BK_LD + f] = tmp[f] * sc;
#pragma unroll
    for (int f = 0; f < DH; ++f) bvF[f * BV_LD + j] = vrow[f];   // feature-major
    if (j < 128) {
      pos_q[j] = pos;
      const int hash = (int)(key >> 17);      // bucket>>5
      tick_q[j] = hash * SEQ + pos;
#pragma unroll
      for (int f = 0; f < DH; ++f) bq[j * BQ_LD + f] = tmp[f];
    }
  }
  __syncthreads();

  const int wave = tid >> 5;
  const int lane = tid & 31;
  const int half = lane >> 4;
  const int r    = lane & 15;
  const int m0   = wave * 16;

  // ---- dots = bq @ bk^T * (1/8), self-mask by original position ----
  for (int n = 0; n < 16; ++n) {
    v8f acc = v8f{};
#pragma unroll
    for (int kt = 0; kt < 16; ++kt) {
      const int kk = kt * 4 + half * 2;
      const v2f a  = *(const v2f*)&bq[(m0 + r) * BQ_LD + kk];
      const v2f bf = *(const v2f*)&bk[(n * 16 + r) * BK_LD + kk];
      acc = __builtin_amdgcn_wmma_f32_16x16x4_f32(
          false, a, false, bf, (short)0, acc, false, false);
    }
    const int col = n * 16 + r;
    const int pk  = pos_kv[col];
#pragma unroll
    for (int i = 0; i < 8; ++i) {
      const int m = m0 + half * 8 + i;
      float dv = acc[i] * 0.125f;            // DH^-0.5
      if (pos_q[m] == pk) dv = SELF_VAL;
      dots[m * DOTS_LD + col] = dv;
    }
  }
  __syncthreads();

  // ---- stable logsumexp per row, write lse (scatter by ticker) ----
  if (tid < 128) {
    const int m = tid;
    float mx = -3.4e38f;
    for (int j = 0; j < 256; ++j) mx = fmaxf(mx, dots[m * DOTS_LD + j]);
    float s = 0.0f;
    for (int j = 0; j < 256; ++j) s += __expf(dots[m * DOTS_LD + j] - mx);
    const float lse = mx + __logf(s);
    lse_s[m] = lse;
    lse_out[(size_t)bh * NKEY + tick_q[m]] = lse;
  }
  __syncthreads();

  // ---- probs = exp(dots - lse), in place ----
  for (int idx = tid; idx < 128 * 256; idx += 256) {
    const int m = idx >> 8, j = idx & 255;
    dots[m * DOTS_LD + j] = __expf(dots[m * DOTS_LD + j] - lse_s[m]);
  }
  __syncthreads();

  // ---- bo = probs @ bv, scatter rows by ticker ----
  for (int n = 0; n < 4; ++n) {
    v8f acc = v8f{};
#pragma unroll
    for (int kt = 0; kt < 64; ++kt) {
      const int kk = kt * 4 + half * 2;
      const v2f a  = *(const v2f*)&dots[(m0 + r) * DOTS_LD + kk];
      const v2f bf = *(const v2f*)&bvF[(n * 16 + r) * BV_LD + kk];
      acc = __builtin_amdgcn_wmma_f32_16x16x4_f32(
          false, a, false, bf, (short)0, acc, false, false);
    }
#pragma unroll
    for (int i = 0; i < 8; ++i) {
      const int m = m0 + half * 8 + i;
      o_hash[((size_t)bh * NKEY + tick_q[m]) * DH + n * 16 + r] = acc[i];
    }
  }
}

// ---------------------------------------------------------------------------
// Combine hash rounds: softmax over per-hash lse, weighted sum, un-merge heads.
// ---------------------------------------------------------------------------
__global__ __launch_bounds__(256) void combine_kernel(
    const float* __restrict__ o_hash,     // (BHN, NKEY, DH)
    const float* __restrict__ lse,        // (BHN, NKEY)
    float* __restrict__ merged) {         // (BT, QKV)
  const int g  = blockIdx.x * 256 + threadIdx.x;
  const int bh = g >> 12;
  const int t  = g & (SEQ - 1);
  const int b_ = bh >> 3;
  const int h  = bh & 7;

  float l[NHASH];
#pragma unroll
  for (int hs = 0; hs < NHASH; ++hs) l[hs] = lse[(size_t)bh * NKEY + hs * SEQ + t];
  float mx = fmaxf(fmaxf(l[0], l[1]), fmaxf(l[2], l[3]));
  float e[NHASH], s = 0.0f;
#pragma unroll
  for (int hs = 0; hs < NHASH; ++hs) { e[hs] = __expf(l[hs] - mx); s += e[hs]; }
  const float inv = 1.0f / s;

  const float* r0 = o_hash + ((size_t)bh * NKEY + 0 * SEQ + t) * DH;
  const float* r1 = o_hash + ((size_t)bh * NKEY + 1 * SEQ + t) * DH;
  const float* r2 = o_hash + ((size_t)bh * NKEY + 2 * SEQ + t) * DH;
  const float* r3 = o_hash + ((size_t)bh * NKEY + 3 * SEQ + t) * DH;
  float* dst = merged + ((size_t)b_ * SEQ + t) * QKV + h * DH;
#pragma unroll
  for (int f = 0; f < DH; ++f)
    dst[f] = (e[0] * r0[f] + e[1] * r1[f] + e[2] * r2[f] + e[3] * r3[f]) * inv;
}

// ---------------------------------------------------------------------------
// Host-side orchestration
// ---------------------------------------------------------------------------
extern "C" void kernel_launch(void* const* d_in, const int* in_sizes, int n_in,
                              void* d_out, int out_size, void* d_ws, size_t ws_size,
                              hipStream_t stream) {
  const float* x        = (const float*)d_in[0];
  const float* w_temp3  = (const float*)d_in[1];
  const float* w_toqk   = (const float*)d_in[2];
  const float* w_tov    = (const float*)d_in[3];
  const float* w_out    = (const float*)d_in[4];
  const float* b_out    = (const float*)d_in[5];
  const float* rotations= (const float*)d_in[6];
  float* out = (float*)d_out;
  float* ws  = (float*)d_ws;

  // workspace layout (floats)
  const size_t OFF_XP   = 0;                          // (BT, EMB) — reused as merged
  const size_t OFF_QK   = OFF_XP + (size_t)BT * EMB;  // (BT, QKV)
  const size_t OFF_V    = OFF_QK + (size_t)BT * QKV;  // (BT, QKV)
  const size_t OFF_KEYS = OFF_V  + (size_t)BT * QKV;  // (BHN, NKEY) u32
  const size_t OFF_OH   = OFF_KEYS + (size_t)BHN * NKEY;   // (BHN, NKEY, DH)
  const size_t OFF_LSE  = OFF_OH + (size_t)BHN * NKEY * DH;

  float* xp      = ws + OFF_XP;
  float* qkb     = ws + OFF_QK;
  float* vb      = ws + OFF_V;
  unsigned* keys = (unsigned*)(ws + OFF_KEYS);
  float* o_hash  = ws + OFF_OH;
  float* lse     = ws + OFF_LSE;
  float* merged  = ws + OFF_XP;   // reuse xp after qk/v are computed

  // 1-3: projections (f32 WMMA GEMMs)
  wmma_gemm_f32_kernel<<<dim3(EMB / 64, BT / 128), 256, 0, stream>>>(
      x, w_temp3, nullptr, xp, BT, EMB, EMB);
  wmma_gemm_f32_kernel<<<dim3(QKV / 64, BT / 128), 256, 0, stream>>>(
      xp, w_toqk, nullptr, qkb, BT, QKV, EMB);
  wmma_gemm_f32_kernel<<<dim3(QKV / 64, BT / 128), 256, 0, stream>>>(
      xp, w_tov, nullptr, vb, BT, QKV, EMB);

  // 4: LSH hashing -> injective sort keys
  lsh_hash_kernel<<<(BHN * SEQ) / 256, 256, 0, stream>>>(qkb, rotations, keys);

  // 5: per-bh bitonic sort of 16384 keys in LDS
  bitonic_sort_kernel<<<BHN, 1024, 0, stream>>>(keys);

  // 6: chunked attention (WMMA), scatter outputs by recovered ticker
  const size_t attn_smem =
      (size_t)(128 * BQ_LD + 256 * BK_LD + 64 * BV_LD + 128 * DOTS_LD) * sizeof(float) +
      (128 + 256 + 128) * sizeof(int) + 128 * sizeof(float);
  lsh_attn_chunk_kernel<<<BHN * NCHUNK, 256, attn_smem, stream>>>(
      qkb, vb, keys, o_hash, lse);

  // 7: combine hash rounds, un-merge heads into `merged`
  combine_kernel<<<(BHN * SEQ) / 256, 256, 0, stream>>>(o_hash, lse, merged);

  // 8: output projection + bias
  wmma_gemm_f32_kernel<<<dim3(EMB / 64, BT / 128), 256, 0, stream>>>(
      merged, w_out, b_out, out, BT, EMB, QKV);
}